// TensoFlow_35923106464323
// MI455X (gfx1250) — compile-verified
//
#include <hip/hip_runtime.h>
#include <hip/hip_bf16.h>

// pwquad_flow for MI455X (gfx1250).
// Streaming workload: 516MB in / 16MB out -> ~23us floor @ 23.3 TB/s.
// Compute is 129 expf/point, so this
// round: native v_exp_f32/v_log_f32, rolled k-loop (3x smaller code), NT
// temporal hint on the async DMA stream (single-use data, don't thrash L2).
// Data path unchanged: wave-private global_load_async_to_lds_b128 (ASYNCcnt),
// s_wait_asynccnt, bank-conflict-free LDS rows (stride 129 dwords, odd).

#define N_PTS        1000000
#define K_DIM        3
#define ROW_DW       129          // 3*43 dwords per point
#define PT_BYTES     516
#define WAVE_PTS     32
#define WAVE_BYTES   16512        // 32*516, multiple of 16
#define WAVE_DW      4128
#define WAVES_PER_BLK 2
#define BLK_THREADS  64
#define N_BLOCKS     15625        // 1e6 / 64, exact

__global__ __launch_bounds__(BLK_THREADS)
void pwquad_kernel(const float* __restrict__ y,
                   const float* __restrict__ wv,
                   float* __restrict__ xout,
                   float* __restrict__ ljout)
{
    __shared__ __align__(16) float lds[WAVES_PER_BLK * WAVE_DW];

    const int lane = threadIdx.x & 31;
    const int wid  = threadIdx.x >> 5;
    const int waveIdx = blockIdx.x * WAVES_PER_BLK + wid;      // 0..31249
    const int p = waveIdx * WAVE_PTS + lane;                   // point id, always < N_PTS

    // --- async stage: this wave's 16512B contiguous region -> its LDS slice ---
    const unsigned ldsWaveByte =
        (unsigned)(size_t)(&lds[0]) + (unsigned)(wid * WAVE_BYTES);
    const unsigned long long gWaveByte =
        (unsigned long long)(size_t)wv + (unsigned long long)waveIdx * WAVE_BYTES;

    #pragma unroll 4
    for (int j = 0; j < 32; ++j) {
        unsigned off = (unsigned)((j * 32 + lane) * 16);       // coalesced 512B/instr
        unsigned la  = ldsWaveByte + off;
        unsigned long long ga = gWaveByte + off;
        asm volatile("global_load_async_to_lds_b128 %0, %1, off th:TH_LOAD_NT"
                     :: "v"(la), "v"(ga) : "memory");
    }
    {   // tail: last 128B (32 dwords) of the 16512B region
        unsigned off = 16384u + (unsigned)(lane * 4);
        unsigned la  = ldsWaveByte + off;
        unsigned long long ga = gWaveByte + off;
        asm volatile("global_load_async_to_lds_b32 %0, %1, off th:TH_LOAD_NT"
                     :: "v"(la), "v"(ga) : "memory");
    }
    __builtin_amdgcn_s_wait_asynccnt(0);   // wave-private region: no barrier needed

    // --- compute: per lane = one point, 3 dims sequentially (rolled loop) ---
    const int rowBase = wid * WAVE_DW + lane * ROW_DW;   // stride 129 (odd) -> conflict-free
    const float EPSF  = 1.1920928955078125e-7f;          // 2^-23 (np.float32 eps)
    const float ONEmE = 1.0f - EPSF;

    float logj = 0.0f;

    #pragma unroll 1
    for (int k = 0; k < K_DIM; ++k) {
        const int db = rowBase + k * 43;
        const float yk = y[(size_t)p * 3 + k];

        // ve = exp(v_tilde[0..21])  (native v_exp_f32 path)
        float ve[22];
        #pragma unroll
        for (int i = 0; i < 22; ++i) ve[i] = __expf(lds[db + i]);

        // w_raw = exp(w_tilde[0..20]); wnorm = sum; traps = sum((ve_i+ve_{i+1})/2 * w_raw)
        float wE[21];
        float wnorm = 0.0f, traps = 0.0f;
        #pragma unroll
        for (int i = 0; i < 21; ++i) {
            float w_ = __expf(lds[db + 22 + i]);
            wE[i] = w_;
            wnorm += w_;
            traps += (ve[i] + ve[i + 1]) * 0.5f * w_;
        }
        // v[i] = max(ve[i] * wnorm/traps, 1e-6);  w[i] = wE[i]/wnorm
        const float invtrap = wnorm / traps;
        const float invw    = 1.0f / wnorm;

        // streaming bin search over monotone vw; capture last bin with vw[i] <= y
        float vw = 0.0f, wcum = 0.0f;
        float vi = fmaxf(ve[0] * invtrap, 1e-6f);
        float e_v0 = vi, e_v1 = 0.0f, e_w = 0.0f, e_c = 0.0f, e_sh = 0.0f;
        #pragma unroll
        for (int i = 0; i < 21; ++i) {
            float vip = fmaxf(ve[i + 1] * invtrap, 1e-6f);
            float wi  = wE[i] * invw;
            bool sel  = (vw <= yk);               // vw here is vw[i]
            e_v0 = sel ? vi   : e_v0;
            e_v1 = sel ? vip  : e_v1;
            e_w  = sel ? wi   : e_w;
            e_c  = sel ? vw   : e_c;              // vw[edge]
            e_sh = sel ? wcum : e_sh;             // wsum_shift[edge]
            vw   += (vi + vip) * 0.5f * wi;
            wcum += wi;
            vi = vip;
        }

        // quadratic solve inside selected bin
        float a  = (e_v1 - e_v0) * e_w;
        float bq = e_v0 * e_w;
        float c  = e_c - yk;
        a = (fabsf(a) < EPSF) ? EPSF : a;
        float dd = fmaxf(bq * bq - 2.0f * a * c, 0.0f);
        float sq = sqrtf(dd);
        float inva = 1.0f / a;
        float s1 = (-bq - sq) * inva;
        float s2 = (-bq + sq) * inva;
        float sol = (s1 >= 0.0f && s1 < 1.0f) ? s1 : s2;
        sol = fminf(fmaxf(sol, EPSF), ONEmE);

        float xv = e_w * sol + e_sh;
        xout[(size_t)p * 3 + k] = fminf(fmaxf(xv, EPSF), ONEmE);

        logj -= __logf(e_v0 + (e_v1 - e_v0) * sol);
    }

    ljout[p] = logj;
}

extern "C" void kernel_launch(void* const* d_in, const int* in_sizes, int n_in,
                              void* d_out, int out_size, void* d_ws, size_t ws_size,
                              hipStream_t stream) {
    (void)in_sizes; (void)n_in; (void)d_ws; (void)ws_size; (void)out_size;
    const float* y  = (const float*)d_in[0];            // (1M, 3) f32
    const float* wv = (const float*)d_in[1];            // (1M, 3, 43) f32
    float* xout = (float*)d_out;                        // first 3M floats: x
    float* lj   = (float*)d_out + (size_t)3 * N_PTS;    // next 1M floats: logj
    pwquad_kernel<<<dim3(N_BLOCKS), dim3(BLK_THREADS), 0, stream>>>(y, wv, xout, lj);
}